// MultiBoxLoss_39178691674621
// MI455X (gfx1250) — compile-verified
//
#include <hip/hip_runtime.h>
#include <cstdint>
#include <cstddef>

#define B_      64
#define P_      24564
#define C_      21
#define G_      24
#define THRESH  0.5f
#define NEGPOS  3
#define VAR0_   0.1f
#define VAR1_   0.2f
#define TPB     256

typedef __attribute__((ext_vector_type(2))) float v2f;
typedef __attribute__((ext_vector_type(8))) float v8f;

// ---------------------------------------------------------------------------
// Wave32 sum-reduction via the CDNA5 matrix pipe (V_WMMA_F32_16X16X4_F32).
// A (16x4) holds one partial per lane (a[1]=0), B = ones -> D[m][n] = v[m] +
// v[m+16]; per-lane sum of the 8 D VGPRs + one xor-16 shuffle = 32-lane sum.
// Requires EXEC == all ones (callers have no early exits before this point).
// ---------------------------------------------------------------------------
__device__ __forceinline__ float wave_sum_wmma(float v) {
    v2f a;  a[0] = v;    a[1] = 0.0f;
    v2f bo; bo[0] = 1.0f; bo[1] = 1.0f;
    v8f c = {};
    c = __builtin_amdgcn_wmma_f32_16x16x4_f32(false, a, false, bo,
                                              (short)0, c, false, false);
    float part = c[0] + c[1] + c[2] + c[3] + c[4] + c[5] + c[6] + c[7];
    part += __shfl_xor(part, 16, 32);
    return part;   // full-wave sum, broadcast to all lanes
}

__device__ __forceinline__ float smooth_l1(float d) {
    float ad = fabsf(d);
    return (ad < 1.0f) ? 0.5f * d * d : ad - 0.5f;
}

// ---------------------------------------------------------------------------
// Kernel 1: per-truth argmax over priors (packed u64 atomic max).
// packed = iou_bits<<32 | (0xFFFFFF - p)  -> ties resolve to smallest p,
// matching jnp.argmax first-index semantics. IoU >= 0 so bit order == order.
// ---------------------------------------------------------------------------
__global__ void kBestPrior(const float4* __restrict__ dbox,
                           const float*  __restrict__ targets,
                           unsigned long long* __restrict__ bestp) {
    const int b = blockIdx.y;
    __shared__ float s_t[G_ * 5];
    __shared__ unsigned long long s_best[G_];
    const int t = threadIdx.x;
    if (t < G_ * 5) s_t[t] = targets[(size_t)b * G_ * 5 + t];
    if (t < G_)     s_best[t] = 0ULL;
    __syncthreads();

    int p  = blockIdx.x * TPB + t;
    int pc = (p < P_) ? p : (P_ - 1);           // duplicate lane is idempotent
    float4 d = dbox[pc];
    float px0 = d.x - 0.5f * d.z, py0 = d.y - 0.5f * d.w;
    float px1 = d.x + 0.5f * d.z, py1 = d.y + 0.5f * d.w;
    float areaB = (px1 - px0) * (py1 - py0);

    #pragma unroll
    for (int g = 0; g < G_; ++g) {
        float x0 = s_t[g * 5 + 0], y0 = s_t[g * 5 + 1];
        float x1 = s_t[g * 5 + 2], y1 = s_t[g * 5 + 3];
        float iw = fmaxf(fminf(px1, x1) - fmaxf(px0, x0), 0.0f);
        float ih = fmaxf(fminf(py1, y1) - fmaxf(py0, y0), 0.0f);
        float inter = iw * ih;
        float areaA = (x1 - x0) * (y1 - y0);
        float iou = inter / (areaA + areaB - inter);
        unsigned long long packed =
            ((unsigned long long)__float_as_uint(iou) << 32) |
            (unsigned long long)(0x00FFFFFFu - (unsigned)pc);
        atomicMax(&s_best[g], packed);          // ds_max_u64
    }
    __syncthreads();
    if (t < G_) atomicMax(&bestp[(size_t)b * G_ + t], s_best[t]); // global_atomic_max_u64
}

// ---------------------------------------------------------------------------
// Kernel 2: match + encode + smooth-L1 + streaming log-softmax CE.
// The block's contiguous 21504-byte category chunk is streamed into LDS with
// per-lane GLOBAL_LOAD_ASYNC_TO_LDS_B128 (ASYNCcnt engine), overlapped with
// the IoU/encode math, drained with s_wait_asynccnt before the softmax.
// ---------------------------------------------------------------------------
__global__ void kLoss(const float4* __restrict__ loc,
                      const float*  __restrict__ cat,
                      const float4* __restrict__ dbox,
                      const float*  __restrict__ targets,
                      const unsigned long long* __restrict__ bestp,
                      float* __restrict__ ce_neg,
                      float* __restrict__ pos_cnt,
                      float* __restrict__ loss_l_acc,
                      float* __restrict__ loss_c_acc) {
    const int b = blockIdx.y;
    __shared__ float s_cat[TPB * C_];           // 21504 B category tile
    __shared__ float s_t[G_ * 5];
    __shared__ int   s_bp[G_];
    const int t = threadIdx.x;

    // ---- kick off async category tile copy (global -> LDS, 16B per lane) ----
    const int blk_start = blockIdx.x * TPB;
    const int npri = min(TPB, P_ - blk_start);
    const int ntr  = (npri * C_) >> 2;          // 16-byte transfers (npri*84/16)
    {
        const unsigned long long gbase =
            (unsigned long long)(const void*)(cat + ((size_t)b * P_ + blk_start) * C_);
        const unsigned lds_base = (unsigned)(size_t)(void*)s_cat; // addr[31:0] = LDS offset
        for (int i = t; i < ntr; i += TPB) {
            unsigned           ldsa = lds_base + 16u * (unsigned)i;
            unsigned long long ga   = gbase + 16ull * (unsigned long long)i;
            asm volatile("global_load_async_to_lds_b128 %0, %1, off"
                         :: "v"(ldsa), "v"(ga) : "memory");
        }
    }

    if (t < G_ * 5) s_t[t] = targets[(size_t)b * G_ * 5 + t];
    if (t < G_)
        s_bp[t] = 0x00FFFFFF - (int)(bestp[(size_t)b * G_ + t] & 0xFFFFFFFFULL);
    __syncthreads();

    const int  p     = blk_start + t;
    const bool valid = (p < P_);
    const int  pc    = valid ? p : (P_ - 1);

    __builtin_prefetch(&loc[(size_t)b * P_ + pc], 0, 3);   // global_prefetch_b8

    float4 d = dbox[pc];
    float px0 = d.x - 0.5f * d.z, py0 = d.y - 0.5f * d.w;
    float px1 = d.x + 0.5f * d.z, py1 = d.y + 0.5f * d.w;
    float areaB = (px1 - px0) * (py1 - py0);

    float ov = -1.0f;
    int   idx = 0;
    #pragma unroll
    for (int g = 0; g < G_; ++g) {              // argmax over truths, first-index ties
        float x0 = s_t[g * 5 + 0], y0 = s_t[g * 5 + 1];
        float x1 = s_t[g * 5 + 2], y1 = s_t[g * 5 + 3];
        float iw = fmaxf(fminf(px1, x1) - fmaxf(px0, x0), 0.0f);
        float ih = fmaxf(fminf(py1, y1) - fmaxf(py0, y0), 0.0f);
        float inter = iw * ih;
        float areaA = (x1 - x0) * (y1 - y0);
        float iou = inter / (areaA + areaB - inter);
        if (iou > ov) { ov = iou; idx = g; }
    }
    #pragma unroll
    for (int g = 0; g < G_; ++g) {              // best-prior override (last g wins)
        if (s_bp[g] == pc) { ov = 2.0f; idx = g; }
    }
    int  conf = (ov < THRESH) ? 0 : ((int)s_t[idx * 5 + 4] + 1);
    bool pos  = conf > 0;

    // encode(matched, prior)
    float mx0 = s_t[idx * 5 + 0], my0 = s_t[idx * 5 + 1];
    float mx1 = s_t[idx * 5 + 2], my1 = s_t[idx * 5 + 3];
    float l0 = ((mx0 + mx1) * 0.5f - d.x) / (VAR0_ * d.z);
    float l1 = ((my0 + my1) * 0.5f - d.y) / (VAR0_ * d.w);
    float l2 = logf((mx1 - mx0) / d.z) / VAR1_;
    float l3 = logf((my1 - my0) / d.w) / VAR1_;

    float4 lv = loc[(size_t)b * P_ + pc];
    float sl = smooth_l1(lv.x - l0) + smooth_l1(lv.y - l1) +
               smooth_l1(lv.z - l2) + smooth_l1(lv.w - l3);
    float ll = (pos && valid) ? sl : 0.0f;

    // ---- drain async copy, then streaming log-softmax CE from LDS ----
    asm volatile("s_wait_asynccnt 0" ::: "memory");
    __syncthreads();

    const float* xs = s_cat + t * C_;           // stride-21 words: conflict-free
    float x[C_];
    #pragma unroll
    for (int c = 0; c < C_; ++c) x[c] = xs[c];
    float m = x[0];
    #pragma unroll
    for (int c = 1; c < C_; ++c) m = fmaxf(m, x[c]);
    float s = 0.0f;
    #pragma unroll
    for (int c = 0; c < C_; ++c) s += expf(x[c] - m);
    float lse = m + logf(s);
    float ce  = lse - x[conf];

    float ce_pos = (pos && valid) ? ce : 0.0f;
    float cn     = (!pos && valid) ? ce : 0.0f;
    if (valid) ce_neg[(size_t)b * P_ + p] = cn;
    float pcf = (pos && valid) ? 1.0f : 0.0f;

    // full-EXEC WMMA wave reductions, one atomic triple per wave
    float rl = wave_sum_wmma(ll);
    float rc = wave_sum_wmma(ce_pos);
    float rp = wave_sum_wmma(pcf);
    if ((t & 31) == 0) {
        atomicAdd(loss_l_acc, rl);
        atomicAdd(loss_c_acc, rc);
        atomicAdd(&pos_cnt[b], rp);
    }
}

// ---------------------------------------------------------------------------
// Kernel 3: hard-negative mining via MSB-first radix select on float bits
// (ce_neg >= 0 so uint order == float order). top-k sum is tie-exact vs. the
// argsort reference: sum(x > T) + (k - count(x > T)) * T.
// ---------------------------------------------------------------------------
__global__ void kNegMine(const float* __restrict__ ce_neg,
                         const float* __restrict__ pos_cnt,
                         float* __restrict__ loss_c_acc,
                         int*   __restrict__ n_total) {
    const int b = blockIdx.x;
    const float* cn = ce_neg + (size_t)b * P_;
    __shared__ unsigned hist[256];
    __shared__ unsigned s_prefix, s_k;
    __shared__ float    s_wsum[TPB / 32];
    __shared__ float    s_wcnt[TPB / 32];

    const int t = threadIdx.x;
    const int num_pos = (int)(pos_cnt[b] + 0.5f);
    const int k = min(NEGPOS * num_pos, P_ - 1);
    if (t == 0) {
        atomicAdd(n_total, num_pos);
        s_prefix = 0u;
        s_k = (unsigned)k;
    }
    __syncthreads();
    if (k <= 0) return;                          // uniform per block

    for (int pass = 0; pass < 4; ++pass) {
        const int shift = 24 - 8 * pass;
        for (int i = t; i < 256; i += TPB) hist[i] = 0u;
        __syncthreads();
        const unsigned prefix = s_prefix;
        for (int i = t; i < P_; i += TPB) {
            unsigned bits = __float_as_uint(cn[i]);
            if (pass == 0 || (bits >> (shift + 8)) == prefix)
                atomicAdd(&hist[(bits >> shift) & 0xFFu], 1u);
        }
        __syncthreads();
        if (t == 0) {
            unsigned kk = s_k, bin = 0;
            for (int bc = 255; bc >= 0; --bc) {
                unsigned c = hist[bc];
                if (kk <= c) { bin = (unsigned)bc; break; }
                kk -= c;
            }
            s_prefix = (prefix << 8) | bin;
            s_k = kk;
        }
        __syncthreads();
    }

    const unsigned T = s_prefix;                 // bits of the k-th largest
    float lsum = 0.0f, lcnt = 0.0f;
    for (int i = t; i < P_; i += TPB) {
        unsigned bits = __float_as_uint(cn[i]);
        if (bits > T) { lsum += cn[i]; lcnt += 1.0f; }
    }
    float wsum = wave_sum_wmma(lsum);            // full EXEC here
    float wcnt = wave_sum_wmma(lcnt);
    if ((t & 31) == 0) { s_wsum[t >> 5] = wsum; s_wcnt[t >> 5] = wcnt; }
    __syncthreads();
    if (t == 0) {
        float sum = 0.0f, cntf = 0.0f;
        for (int w = 0; w < TPB / 32; ++w) { sum += s_wsum[w]; cntf += s_wcnt[w]; }
        float ans = sum + (float)(k - (int)(cntf + 0.5f)) * __uint_as_float(T);
        atomicAdd(loss_c_acc, ans);
    }
}

__global__ void kFinal(const float* __restrict__ loss_l,
                       const float* __restrict__ loss_c,
                       const int*   __restrict__ n_total,
                       float* __restrict__ out) {
    float N = (float)(*n_total);
    out[0] = *loss_l / N;
    out[1] = *loss_c / N;
}

// ---------------------------------------------------------------------------
// Workspace layout (bytes):
//   [0,      12288)  u64 best_prior[B*G]          (zeroed)
//   [12288,  12544)  f32 pos_cnt[B]               (zeroed)
//   [12544]          f32 loss_l accumulator       (zeroed)
//   [12548]          f32 loss_c accumulator       (zeroed)
//   [12552]          i32 n_total                  (zeroed)
//   [12608,  ...)    f32 ce_neg[B*P]              (fully overwritten)
// ---------------------------------------------------------------------------
extern "C" void kernel_launch(void* const* d_in, const int* in_sizes, int n_in,
                              void* d_out, int out_size, void* d_ws, size_t ws_size,
                              hipStream_t stream) {
    (void)in_sizes; (void)n_in; (void)out_size; (void)ws_size;
    const float4* loc     = (const float4*)d_in[0];
    const float*  cat     = (const float*) d_in[1];
    const float4* dbox    = (const float4*)d_in[2];
    const float*  targets = (const float*) d_in[3];
    float* out = (float*)d_out;

    unsigned char* ws = (unsigned char*)d_ws;
    unsigned long long* bestp   = (unsigned long long*)ws;
    float* pos_cnt    = (float*)(ws + 12288);
    float* loss_l_acc = (float*)(ws + 12544);
    float* loss_c_acc = (float*)(ws + 12548);
    int*   n_total    = (int*)  (ws + 12552);
    float* ce_neg     = (float*)(ws + 12608);

    hipMemsetAsync(d_ws, 0, 12608, stream);

    dim3 grid((P_ + TPB - 1) / TPB, B_);
    kBestPrior<<<grid, TPB, 0, stream>>>(dbox, targets, bestp);
    kLoss<<<grid, TPB, 0, stream>>>(loc, cat, dbox, targets, bestp,
                                    ce_neg, pos_cnt, loss_l_acc, loss_c_acc);
    kNegMine<<<B_, TPB, 0, stream>>>(ce_neg, pos_cnt, loss_c_acc, n_total);
    kFinal<<<1, 1, 0, stream>>>(loss_l_acc, loss_c_acc, n_total, out);
}